// RandomRoll_54365696033616
// MI455X (gfx1250) — compile-verified
//
#include <hip/hip_runtime.h>

#ifndef __has_builtin
#define __has_builtin(x) 0
#endif

#define BATCH 64
#define CHAN 3
#define HDIM 512
#define WDIM 512

typedef __attribute__((ext_vector_type(4))) int v4i;
typedef __attribute__((address_space(1))) v4i GlobV4i;   // global (AS1)
typedef __attribute__((address_space(3))) v4i LdsV4i;    // LDS (AS3)

// One block per output row (b, c, h): 128 threads x 4 floats = 512-float row.
// Global->LDS staged copy (async-to-LDS on CDNA5), rotation resolved in LDS,
// aligned 128-bit stores back to global.
__global__ __launch_bounds__(128) void RandomRoll_rows_kernel(
    const float* __restrict__ x,
    const int*   __restrict__ shifts,
    float*       __restrict__ out) {
  __shared__ float row[WDIM];

  const int t = threadIdx.x;   // 0..127
  const int h = blockIdx.x;    // 0..511
  const int c = blockIdx.y;    // 0..2
  const int b = blockIdx.z;    // 0..63

  // Uniform per block -> scalar loads
  const int s0 = shifts[2 * b + 0];
  const int s1 = shifts[2 * b + 1];

  const int h_in = (h - s0) & (HDIM - 1);
  const size_t plane = ((size_t)b * CHAN + c) * (size_t)(HDIM * WDIM);
  const float* __restrict__ src_row = x   + plane + (size_t)h_in * WDIM;
  float*       __restrict__ dst_row = out + plane + (size_t)h    * WDIM;

#if __has_builtin(__builtin_amdgcn_global_load_async_to_lds_b128) && \
    __has_builtin(__builtin_amdgcn_s_wait_asynccnt)
  {
    // CDNA5 async copy: each lane moves 16B directly into LDS (ASYNCcnt).
    GlobV4i* g = (GlobV4i*)(src_row + 4 * t);   // C-style cast: const + AS cast
    LdsV4i*  l = (LdsV4i*)(&row[4 * t]);
    __builtin_amdgcn_global_load_async_to_lds_b128(g, l, /*offset=*/0, /*cpol=*/0);
    __builtin_amdgcn_s_wait_asynccnt(0);
  }
#else
  // Fallback: aligned 128-bit load/store through VGPRs.
  *(float4*)(&row[4 * t]) = *(const float4*)(src_row + 4 * t);
#endif
  __syncthreads();

  // Rotate within the row via LDS gather, then aligned 128-bit store.
  const int wb = 4 * t;
  float4 o;
  o.x = row[(wb + 0 - s1) & (WDIM - 1)];
  o.y = row[(wb + 1 - s1) & (WDIM - 1)];
  o.z = row[(wb + 2 - s1) & (WDIM - 1)];
  o.w = row[(wb + 3 - s1) & (WDIM - 1)];
  *(float4*)(dst_row + wb) = o;
}

extern "C" void kernel_launch(void* const* d_in, const int* in_sizes, int n_in,
                              void* d_out, int out_size, void* d_ws, size_t ws_size,
                              hipStream_t stream) {
  (void)in_sizes; (void)n_in; (void)out_size; (void)d_ws; (void)ws_size;
  const float* x      = (const float*)d_in[0];   // [64,3,512,512] fp32
  const int*   shifts = (const int*)d_in[1];     // [64,2] int32
  float*       out    = (float*)d_out;

  dim3 grid(HDIM, CHAN, BATCH);   // (h, c, b)
  dim3 block(128, 1, 1);          // 4 waves of 32
  RandomRoll_rows_kernel<<<grid, block, 0, stream>>>(x, shifts, out);
}